// LearnableMidiSynth_13872744366719
// MI455X (gfx1250) — compile-verified
//
#include <hip/hip_runtime.h>
#include <math.h>

typedef __attribute__((ext_vector_type(2))) float v2f;
typedef __attribute__((ext_vector_type(8))) float v8f;

#define SYN_L     32768
#define SYN_W     1024
#define SYN_H     8
#define SYN_K     128
#define TILE      512              // time samples per workgroup chunk
#define XROWS     (TILE + 128)     // chunk + FIR history
#define NBLK      (TILE / 16)      // 16-wide output blocks per chunk

// ---------------------------------------------------------------------------
// Zero the output buffer (atomic overlap-add accumulates into it each call).
// ---------------------------------------------------------------------------
__global__ __launch_bounds__(256) void zero_kernel(float* __restrict__ out, int n) {
    int i = blockIdx.x * 256 + threadIdx.x;
    if (i < n) out[i] = 0.0f;
}

// ---------------------------------------------------------------------------
// Phase 1: per-note tiny MLPs -> harmonic amplitudes amps[N][8] in workspace.
//   time_embedder: 1 -> 32 (ReLU) -> 2
//   synth MLP:     4 -> 64 (ReLU) -> 8, softplus
// ---------------------------------------------------------------------------
__global__ __launch_bounds__(256) void note_params_kernel(
    const float* __restrict__ freq, const float* __restrict__ vel,
    const float* __restrict__ w1,   const float* __restrict__ b1,
    const float* __restrict__ w2,   const float* __restrict__ b2,
    const float* __restrict__ ws1,  const float* __restrict__ bs1,
    const float* __restrict__ ws2,  const float* __restrict__ bs2,
    const int*   __restrict__ starts,
    float* __restrict__ ampsOut, int N, int Dn)
{
    int n = blockIdx.x * 256 + threadIdx.x;
    if (n >= N) return;

    float nt = (float)starts[n] / (float)Dn;
    float lat0 = b2[0], lat1 = b2[1];
    #pragma unroll 4
    for (int j = 0; j < 32; ++j) {
        float h = fmaf(nt, w1[j], b1[j]);
        h = fmaxf(h, 0.0f);
        lat0 = fmaf(h, w2[j * 2 + 0], lat0);
        lat1 = fmaf(h, w2[j * 2 + 1], lat1);
    }

    float f0 = freq[n];
    float f1 = vel[n] * (1.0f / 127.0f);

    float acc[SYN_H];
    #pragma unroll
    for (int h = 0; h < SYN_H; ++h) acc[h] = bs2[h];

    for (int j = 0; j < 64; ++j) {
        float s = bs1[j];
        s = fmaf(f0,   ws1[0 * 64 + j], s);
        s = fmaf(f1,   ws1[1 * 64 + j], s);
        s = fmaf(lat0, ws1[2 * 64 + j], s);
        s = fmaf(lat1, ws1[3 * 64 + j], s);
        s = fmaxf(s, 0.0f);
        #pragma unroll
        for (int h = 0; h < SYN_H; ++h)
            acc[h] = fmaf(s, ws2[j * SYN_H + h], acc[h]);
    }

    #pragma unroll
    for (int h = 0; h < SYN_H; ++h) {
        float x = acc[h];
        ampsOut[n * SYN_H + h] = (x > 20.0f) ? x : log1pf(__expf(x));  // softplus
    }
}

// ---------------------------------------------------------------------------
// Phase 2 (fused): additive synthesis -> hann window/mask -> 128-tap FIR via
// WMMA (Toeplitz filter blocks x 16 notes) -> tanh -> atomic overlap-add.
// Grid: (N/16 note groups, L/TILE chunks), 256 threads (8 wave32) per WG.
// ---------------------------------------------------------------------------
__global__ __launch_bounds__(256) void synth_fir_kernel(
    const float* __restrict__ freq_rad, const float* __restrict__ velocity,
    const float* __restrict__ fir,      const float* __restrict__ amps,
    const int*   __restrict__ starts,   const int* __restrict__ lengths,
    float* __restrict__ out, int Dn)
{
    __shared__ float sF[9][16][17];       // Toeplitz filter blocks (padded: no bank conflicts)
    __shared__ float sX[XROWS][16];       // windowed signal, time-major [time][note]
    __shared__ float sFreq[16], sVel[16], sAmps[16][SYN_H];
    __shared__ int   sLen[16], sStart[16];

    const int tid   = threadIdx.x;
    const int group = blockIdx.x;         // 16-note group
    const int chunk = blockIdx.y;         // TILE-sample chunk within note

    // ---- per-note parameters -------------------------------------------------
    if (tid < 16) {
        int note = group * 16 + tid;
        sFreq[tid] = freq_rad[note];
        sVel[tid]  = velocity[note] * (1.0f / 127.0f);
        int st = starts[note];
        int ln = lengths[note];
        int ol = min(st + ln, Dn) - st;
        ol = max(0, min(ol, SYN_L));
        sLen[tid]   = ol;
        sStart[tid] = st;
    }
    if (tid < 128) {
        int n = tid >> 3, h = tid & 7;
        sAmps[n][h] = amps[(group * 16 + n) * SYN_H + h];
    }
    // ---- Toeplitz filter blocks: F_d[m][i] = fir[16d + m - i] ---------------
    {
        int m = tid >> 4, i = tid & 15;
        #pragma unroll
        for (int d = 0; d < 9; ++d) {
            int k = 16 * d + m - i;
            sF[d][m][i] = (k >= 0 && k < SYN_K) ? fir[k] : 0.0f;
        }
    }
    __syncthreads();

    // ---- additive synthesis + hann window + length mask into LDS ------------
    // 2 trans ops/sample (sincos) + Chebyshev harmonic recurrence.
    for (int e = tid; e < XROWS * 16; e += 256) {
        int xi = e >> 4, n = e & 15;
        int t  = chunk * TILE + xi - 128;           // global sample index in note
        float v = 0.0f;
        int ol = sLen[n];
        if (t >= 0 && t < ol) {
            float p  = sFreq[n] * (float)t;
            float s1 = __sinf(p);
            float c1 = __cosf(p);
            float twoc = 2.0f * c1;
            float sPrev = 0.0f, sCur = s1, acc = 0.0f;
            #pragma unroll
            for (int h = 0; h < SYN_H; ++h) {
                acc = fmaf(sAmps[n][h], sCur, acc);
                float sNext = fmaf(twoc, sCur, -sPrev); // sin((h+2)p)
                sPrev = sCur; sCur = sNext;
            }
            v = acc * sVel[n];
            int wpos = t - (ol - SYN_W);            // hann over last W samples
            if (wpos >= 0) {                        // wpos < W since t < ol
                v *= 0.5f * (1.0f - __cosf(6.2831853071795864f *
                                           (1.0f / (float)SYN_W) * (float)wpos));
            }
        }
        sX[xi][n] = v;
    }
    __syncthreads();

    // ---- FIR as 9 Toeplitz-block matmuls on V_WMMA_F32_16X16X4_F32 ----------
    const int lane = tid & 31;
    const int wave = tid >> 5;
    const int row  = lane & 15;           // A-row m / B-col note / D-col note
    const int hi   = lane >> 4;           // lane-half selects K pair & M half

    for (int Tb = wave; Tb < NBLK; Tb += 8) {
        v8f c = {};
        for (int d = 0; d < 9; ++d) {
            int xb = 16 * Tb - 16 * d + 128;   // X row base for this tap block
            #pragma unroll
            for (int s4 = 0; s4 < 4; ++s4) {   // chain K=4 slices -> K=16
                int ka = 4 * s4 + 2 * hi;
                v2f A, B;
                A.x = sF[d][row][ka];
                A.y = sF[d][row][ka + 1];
                B.x = sX[xb + ka][row];
                B.y = sX[xb + ka + 1][row];
                c = __builtin_amdgcn_wmma_f32_16x16x4_f32(
                        /*neg_a=*/false, A, /*neg_b=*/false, B,
                        /*c_mod=*/(short)0, c,
                        /*reuse_a=*/false, /*reuse_b=*/false);
            }
        }
        // ---- tanh + masked overlap-add scatter ------------------------------
        int ol = sLen[row];
        int st = sStart[row];
        #pragma unroll
        for (int r = 0; r < 8; ++r) {
            int m = r + 8 * hi;                   // D row per ISA C/D layout
            int t = chunk * TILE + Tb * 16 + m;
            if (t < ol) {
                float y = tanhf(c[r]);
                int idx = st + t;
                if (idx < Dn) atomicAdd(&out[idx], y);
            }
        }
    }
}

// ---------------------------------------------------------------------------
extern "C" void kernel_launch(void* const* d_in, const int* in_sizes, int n_in,
                              void* d_out, int out_size, void* d_ws, size_t ws_size,
                              hipStream_t stream) {
    const float* freq    = (const float*)d_in[0];
    const float* vel     = (const float*)d_in[1];
    const float* w1      = (const float*)d_in[2];
    const float* b1      = (const float*)d_in[3];
    const float* w2      = (const float*)d_in[4];
    const float* b2      = (const float*)d_in[5];
    const float* ws1     = (const float*)d_in[6];
    const float* bs1     = (const float*)d_in[7];
    const float* ws2     = (const float*)d_in[8];
    const float* bs2     = (const float*)d_in[9];
    const float* fir     = (const float*)d_in[10];
    const int*   starts  = (const int*)d_in[11];
    const int*   lengths = (const int*)d_in[12];
    // d_in[13] = duration_samples; equals out_size, use host value.

    float* out  = (float*)d_out;
    float* amps = (float*)d_ws;            // N*8 floats = 32 KB
    int N  = in_sizes[0];
    int Dn = out_size;

    zero_kernel<<<(out_size + 255) / 256, 256, 0, stream>>>(out, out_size);
    note_params_kernel<<<(N + 255) / 256, 256, 0, stream>>>(
        freq, vel, w1, b1, w2, b2, ws1, bs1, ws2, bs2, starts, amps, N, Dn);

    dim3 grid(N / 16, SYN_L / TILE);       // 64 note groups x 64 chunks
    synth_fir_kernel<<<grid, 256, 0, stream>>>(
        freq, vel, fir, amps, starts, lengths, out, Dn);
}